// PDN_dense_add_skip_6bro_pre_batch_act_max_test_38010460569934
// MI455X (gfx1250) — compile-verified
//
#include <hip/hip_runtime.h>

#define NCH 64
#define FE 16
#define BN_EPS 1e-5f

typedef float v2f __attribute__((ext_vector_type(2)));
typedef float v8f __attribute__((ext_vector_type(8)));

__device__ __forceinline__ void atomAddF(float* p, float v) {
    unsafeAtomicAdd(p, v);   // native global_atomic_add_f32 (L2-resident buffers)
}

// ---------------- elementwise helpers ----------------
__global__ __launch_bounds__(256) void fill_kernel(float* __restrict__ p, float v, int n) {
    int i = blockIdx.x * 256 + threadIdx.x;
    if (i < n) p[i] = v;
}

__global__ __launch_bounds__(256) void copy_kernel(const float* __restrict__ s,
                                                   float* __restrict__ d, int n) {
    int i = blockIdx.x * 256 + threadIdx.x;
    if (i < n) d[i] = s[i];
}

// cur += skip; skip += new_cur   (dense additive skip bookkeeping)
__global__ __launch_bounds__(256) void skip_update_kernel(float* __restrict__ cur,
                                                          float* __restrict__ skip, int n) {
    int i = blockIdx.x * 256 + threadIdx.x;
    if (i < n) {
        float c = cur[i] + skip[i];
        cur[i] = c;
        skip[i] += c;
    }
}

// ---------------- edge MLP: ew = sigmoid(relu(ea@W1+b1)@W2+b2); deg[dst]+=ew ----------------
__global__ __launch_bounds__(256) void edge_mlp_kernel(
    const float* __restrict__ ea, const float* __restrict__ W1, const float* __restrict__ b1,
    const float* __restrict__ W2, const float* __restrict__ b2,
    const int* __restrict__ dst, float* __restrict__ ew, float* __restrict__ deg, int E) {
    __shared__ float sW1[FE * FE];
    __shared__ float sb1[FE];
    __shared__ float sW2[FE];
    __shared__ float sb2;
    if (threadIdx.x < FE * FE) sW1[threadIdx.x] = W1[threadIdx.x];
    if (threadIdx.x < FE) { sb1[threadIdx.x] = b1[threadIdx.x]; sW2[threadIdx.x] = W2[threadIdx.x]; }
    if (threadIdx.x == 0) sb2 = b2[0];
    __syncthreads();

    int e = blockIdx.x * 256 + threadIdx.x;
    if (e >= E) return;

    float a[FE];
    const float4* ea4 = (const float4*)(ea + (size_t)e * FE);
#pragma unroll
    for (int q = 0; q < 4; ++q) {
        float4 v = ea4[q];
        a[4 * q + 0] = v.x; a[4 * q + 1] = v.y; a[4 * q + 2] = v.z; a[4 * q + 3] = v.w;
    }
    float z = sb2;
#pragma unroll
    for (int i = 0; i < FE; ++i) {
        float t = sb1[i];
#pragma unroll
        for (int k = 0; k < FE; ++k) t = fmaf(a[k], sW1[k * FE + i], t);
        z = fmaf(fmaxf(t, 0.f), sW2[i], z);
    }
    float w = 1.f / (1.f + expf(-z));
    ew[e] = w;
    atomAddF(&deg[dst[e]], w);
}

__global__ __launch_bounds__(256) void deg_to_dis_kernel(float* __restrict__ d, int n) {
    int i = blockIdx.x * 256 + threadIdx.x;
    if (i < n) d[i] = rsqrtf(d[i]);   // deg >= 1 (self-loop) so always valid
}

// ---------------- batch-norm column stats ----------------
__global__ __launch_bounds__(256) void bn_stats_kernel(const float* __restrict__ X,
                                                       float* __restrict__ stats, int N) {
    int t = blockIdx.x * 256 + threadIdx.x;
    int c = t & (NCH - 1);
    int r = t >> 6;
    int rstride = (gridDim.x * 256) >> 6;
    float s = 0.f, s2 = 0.f;
    for (; r < N; r += rstride) {
        float v = X[(size_t)r * NCH + c];
        s += v;
        s2 = fmaf(v, v, s2);
    }
    atomAddF(&stats[c], s);
    atomAddF(&stats[NCH + c], s2);
}

__global__ void bn_finalize_kernel(float* __restrict__ stats, int N) {
    int c = threadIdx.x;
    if (c < NCH) {
        float invn = 1.0f / (float)N;
        float mu = stats[c] * invn;
        float var = stats[NCH + c] * invn - mu * mu;
        stats[2 * NCH + c] = mu;
        stats[3 * NCH + c] = rsqrtf(var + BN_EPS);
    }
}

// ---------------- H = bn_relu?(X) @ W via V_WMMA_F32_16X16X4_F32 ----------------
// 256 threads = 8 waves; wave w computes rows [blk*128 + 16w, +16) x all 64 cols.
// W staged in LDS K-pair-interleaved so each B fragment is ONE aligned ds_load_b64
// into an even VGPR pair (no packing movs). USE_BN templated: branch-free inner loop.
// No early return: EXEC must be all-ones for WMMA; OOB rows are clamped/zeroed.
template <int USE_BN>
__global__ __launch_bounds__(256) void gemm_wmma_kernel(
    const float* __restrict__ X, const float* __restrict__ W,
    const float* __restrict__ stats,
    float* __restrict__ H, int nrows) {
    // BsI[(k>>1)*128 + n*2 + (k&1)] = W[k*64 + n]  -> fragment (B[ka][n],B[ka+1][n]) contiguous
    __shared__ float BsI[NCH * NCH];
    __shared__ float sMu[NCH];
    __shared__ float sInv[NCH];
    for (int i = threadIdx.x; i < NCH * NCH; i += 256) {
        int k = i >> 6, n = i & (NCH - 1);
        BsI[(k >> 1) * 128 + n * 2 + (k & 1)] = W[i];
    }
    if (USE_BN && threadIdx.x < NCH) {
        sMu[threadIdx.x]  = stats[2 * NCH + threadIdx.x];
        sInv[threadIdx.x] = stats[3 * NCH + threadIdx.x];
    }
    __syncthreads();

    const int wave = threadIdx.x >> 5;
    const int lane = threadIdx.x & 31;
    const int m    = lane & 15;
    const int half = lane >> 4;
    const int row0 = blockIdx.x * 128 + wave * 16;
    const int row  = row0 + m;
    const int rowc = (row < nrows) ? row : (nrows - 1);
    const float keep = (row < nrows) ? 1.f : 0.f;
    const float* xr = X + (size_t)rowc * NCH;

    v8f acc[4];
#pragma unroll
    for (int t = 0; t < 4; ++t)
#pragma unroll
        for (int r = 0; r < 8; ++r) acc[t][r] = 0.f;

#pragma unroll
    for (int k0 = 0; k0 < NCH; k0 += 4) {
        // A fragment: 16x4 tile; lanes 0-15 hold K=k0..k0+1, lanes 16-31 hold K=k0+2..k0+3
        const int ka = k0 + half * 2;
        v2f xa = *(const v2f*)(xr + ka);
        v2f av;
        if (USE_BN) {
            av.x = fmaxf((xa.x - sMu[ka])     * sInv[ka],     0.f) * keep;
            av.y = fmaxf((xa.y - sMu[ka + 1]) * sInv[ka + 1], 0.f) * keep;
        } else {
            av.x = xa.x * keep;
            av.y = xa.y * keep;
        }
        const float* brow = &BsI[(ka >> 1) * 128 + m * 2];
#pragma unroll
        for (int t = 0; t < 4; ++t) {
            v2f bv = *(const v2f*)(brow + t * 32);   // one ds_load_b64 per fragment
            acc[t] = __builtin_amdgcn_wmma_f32_16x16x4_f32(
                false, av, false, bv, (short)0, acc[t], false, false);
        }
    }

    // C/D layout: VGPR r -> rows (row0+r) for lanes 0-15, (row0+8+r) for lanes 16-31
#pragma unroll
    for (int t = 0; t < 4; ++t)
#pragma unroll
        for (int r = 0; r < 8; ++r) {
            int rr = row0 + half * 8 + r;
            if (rr < nrows) H[(size_t)rr * NCH + t * 16 + m] = acc[t][r];
        }
}

// ---------------- OUT = h*dis^2 (self-loop) + bias ----------------
__global__ __launch_bounds__(256) void init_out_kernel(const float* __restrict__ H,
                                                       const float* __restrict__ dis,
                                                       const float* __restrict__ bias,
                                                       float* __restrict__ OUT, int N) {
    int i = blockIdx.x * 256 + threadIdx.x;
    if (i < N * NCH) {
        int n = i >> 6, c = i & (NCH - 1);
        float d = dis[n];
        OUT[i] = fmaf(H[i], d * d, bias[c]);
    }
}

// ---------------- edge aggregation: OUT[dst] += H[src] * dis[s]*ew*dis[d] ----------------
// 16 threads per edge, float4 vectorized; atomics land in L2 (buffers << 192MB L2).
__global__ __launch_bounds__(256) void scatter_kernel(
    const float* __restrict__ H, const int* __restrict__ src, const int* __restrict__ dst,
    const float* __restrict__ ew, const float* __restrict__ dis,
    float* __restrict__ OUT, int E) {
    int t = blockIdx.x * 256 + threadIdx.x;
    int e = t >> 4;
    if (e >= E) return;
    int c = (t & 15) << 2;
    int s = src[e], d = dst[e];
    float nrm = dis[s] * ew[e] * dis[d];
    float4 hv = *(const float4*)(H + (size_t)s * NCH + c);
    float* o = OUT + (size_t)d * NCH + c;
    atomAddF(o + 0, hv.x * nrm);
    atomAddF(o + 1, hv.y * nrm);
    atomAddF(o + 2, hv.z * nrm);
    atomAddF(o + 3, hv.w * nrm);
}

// ---------------- relu + segment-max pool (bit-cast max valid: values >= 0) ----------------
__global__ __launch_bounds__(256) void pool_kernel(const float* __restrict__ cur,
                                                   const int* __restrict__ batch,
                                                   int* __restrict__ pooled, int N) {
    int i = blockIdx.x * 256 + threadIdx.x;
    if (i < N * NCH) {
        int n = i >> 6, c = i & (NCH - 1);
        float v = fmaxf(cur[i], 0.f);
        atomicMax(pooled + (size_t)batch[n] * NCH + c, __float_as_int(v));
    }
}

__global__ __launch_bounds__(256) void final_linear_kernel(
    const float* __restrict__ pooled, const float* __restrict__ linW,
    const float* __restrict__ linb, float* __restrict__ out, int G, int ncls) {
    int i = blockIdx.x * 256 + threadIdx.x;
    if (i >= G * ncls) return;
    int g = i / ncls, c = i % ncls;
    float s = linb[c];
#pragma unroll 16
    for (int k = 0; k < NCH; ++k) s = fmaf(pooled[(size_t)g * NCH + k], linW[k * ncls + c], s);
    out[i] = s;
}

// ---------------- driver ----------------
extern "C" void kernel_launch(void* const* d_in, const int* in_sizes, int n_in,
                              void* d_out, int out_size, void* d_ws, size_t ws_size,
                              hipStream_t stream) {
    const float* x        = (const float*)d_in[0];
    const int*   eidx     = (const int*)d_in[1];
    const int*   batch    = (const int*)d_in[2];
    const float* edge_attr= (const float*)d_in[4];
    const float* Wlin1    = (const float*)d_in[5];
    const float* bias1    = (const float*)d_in[6];
    const float* m1W1     = (const float*)d_in[7];
    const float* m1b1     = (const float*)d_in[8];
    const float* m2W1     = (const float*)d_in[9];
    const float* m2b1     = (const float*)d_in[10];
    const float* hWlin    = (const float*)d_in[11];
    const float* hbias    = (const float*)d_in[12];
    const float* hm1W     = (const float*)d_in[13];
    const float* hm1b     = (const float*)d_in[14];
    const float* hm2W     = (const float*)d_in[15];
    const float* hm2b     = (const float*)d_in[16];
    const float* linW     = (const float*)d_in[17];
    const float* linb     = (const float*)d_in[18];

    const int N    = in_sizes[0] / NCH;
    const int E    = in_sizes[1] / 2;
    const int NCLS = in_sizes[18];
    const int G    = out_size / NCLS;
    const int NH   = in_sizes[12] / NCH;   // 12 hidden convs
    const int NE64 = N * NCH;

    const int* srcp = eidx;
    const int* dstp = eidx + E;

    // workspace layout (floats): ew[E] | dis[N] | h[N*64] | bufA[N*64] | bufB[N*64] | skip[N*64] | stats[256] | pooled[G*64]
    float* ws     = (float*)d_ws;
    float* ew     = ws;
    float* dis    = ew + E;
    float* h      = dis + N;
    float* bufA   = h + (size_t)NE64;
    float* bufB   = bufA + (size_t)NE64;
    float* skip   = bufB + (size_t)NE64;
    float* stats  = skip + (size_t)NE64;
    float* pooled = stats + 256;

    float* cur = bufA;
    float* nxt = bufB;

    auto g1 = [](int n) { return (n + 255) / 256; };

    for (int j = 0; j <= NH; ++j) {
        const float* W1 = (j == 0) ? m1W1  : hm1W  + (size_t)(j - 1) * FE * FE;
        const float* b1 = (j == 0) ? m1b1  : hm1b  + (size_t)(j - 1) * FE;
        const float* W2 = (j == 0) ? m2W1  : hm2W  + (size_t)(j - 1) * FE;
        const float* b2 = (j == 0) ? m2b1  : hm2b  + (size_t)(j - 1);
        const float* Wl = (j == 0) ? Wlin1 : hWlin + (size_t)(j - 1) * NCH * NCH;
        const float* bl = (j == 0) ? bias1 : hbias + (size_t)(j - 1) * NCH;

        // edge weights + gcn_norm (deg init = 1.0 for self-loop)
        fill_kernel<<<g1(N), 256, 0, stream>>>(dis, 1.0f, N);
        edge_mlp_kernel<<<g1(E), 256, 0, stream>>>(edge_attr, W1, b1, W2, b2, dstp, ew, dis, E);
        deg_to_dis_kernel<<<g1(N), 256, 0, stream>>>(dis, N);

        // node transform (fused bn_relu for hidden layers) via WMMA
        if (j == 0) {
            gemm_wmma_kernel<0><<<(N + 127) / 128, 256, 0, stream>>>(x, Wl, stats, h, N);
        } else {
            fill_kernel<<<1, 256, 0, stream>>>(stats, 0.f, 2 * NCH);
            bn_stats_kernel<<<512, 256, 0, stream>>>(cur, stats, N);
            bn_finalize_kernel<<<1, 64, 0, stream>>>(stats, N);
            gemm_wmma_kernel<1><<<(N + 127) / 128, 256, 0, stream>>>(cur, Wl, stats, h, N);
        }

        // aggregation: self-loop + bias, then edge scatter
        init_out_kernel<<<g1(NE64), 256, 0, stream>>>(h, dis, bl, nxt, N);
        scatter_kernel<<<g1(E * 16), 256, 0, stream>>>(h, srcp, dstp, ew, dis, nxt, E);

        float* tmp = cur; cur = nxt; nxt = tmp;

        if (j == 0) {
            copy_kernel<<<g1(NE64), 256, 0, stream>>>(cur, skip, NE64);   // skip_sum = x0
        } else if ((j & 1) == 0) {
            skip_update_kernel<<<g1(NE64), 256, 0, stream>>>(cur, skip, NE64); // block end
        }
    }

    // relu -> per-graph max pool -> readout linear
    fill_kernel<<<g1(G * NCH), 256, 0, stream>>>(pooled, 0.f, G * NCH);
    pool_kernel<<<g1(NE64), 256, 0, stream>>>(cur, batch, (int*)pooled, N);
    final_linear_kernel<<<g1(G * NCLS), 256, 0, stream>>>(pooled, linW, linb, (float*)d_out, G, NCLS);
}